// Model_hydro_lstm_25683904430871
// MI455X (gfx1250) — compile-verified
//
#include <hip/hip_runtime.h>
#include <hip/hip_bf16.h>

typedef float v2f __attribute__((ext_vector_type(2)));
typedef float v8f __attribute__((ext_vector_type(8)));

#define T_DIM  16384
#define IN_DIM 1024
#define S_DIM  512
#define NCOL   2048   // 4 gates * 512 states

// ---------------------------------------------------------------------------
// Kernel 0: transpose weight_input (S,4,I) -> Bt[I][NCOL], Bt[i][g*512+s] = W[s][g][i]
// K-major layout so GEMM B loads are contiguous across lanes.
// ---------------------------------------------------------------------------
__global__ void wtrans_kernel(const float* __restrict__ w_in, float* __restrict__ bt) {
    const int tid = blockIdx.x * blockDim.x + threadIdx.x;   // tid = i*NCOL + n
    const int n = tid & (NCOL - 1);
    const int i = tid >> 11;
    const int s = n & (S_DIM - 1);
    const int g = n >> 9;
    bt[tid] = w_in[((s << 2) + g) * IN_DIM + i];
}

// ---------------------------------------------------------------------------
// Kernel 1: u[t][n] = sum_i x[t][i] * Bt[i][n] via V_WMMA_F32_16X16X4_F32.
// Register-blocked: each wave owns a 32x32 output tile = 2 A-fragments x
// 2 B-fragments x 4 accumulators -> 4 WMMAs per K=4 step from 6 loads
// (1.5 vmem per wmma instead of 3).
// A layout (16x4 f32): lanes 0-15 hold {K=k,K=k+1}, lanes 16-31 hold {K=k+2,K=k+3}.
// C/D layout: VGPR r -> M = r + 8*(lane>=16), N = lane&15.
// ---------------------------------------------------------------------------
__global__ void __launch_bounds__(256)
gemm_u_kernel(const float* __restrict__ x, const float* __restrict__ bt,
              float* __restrict__ u) {
    const int lane = threadIdx.x & 31;
    const int wave = threadIdx.x >> 5;
    const int half = lane >> 4;       // 0: lanes 0-15, 1: lanes 16-31
    const int l16  = lane & 15;
    const int row0 = blockIdx.y << 5;                    // 32-row M tile base (t)
    const int col0 = (blockIdx.x << 8) + (wave << 5);    // 32-col N tile base

    const float* aRow0 = x + (size_t)(row0 + l16) * IN_DIM + 2 * half;        // M  0..15
    const float* aRow1 = aRow0 + (size_t)16 * IN_DIM;                         // M 16..31
    const float* bCol0 = bt + (size_t)(2 * half) * NCOL + col0 + l16;         // N  0..15
    const float* bCol1 = bCol0 + 16;                                          // N 16..31

    v8f acc00 = {0.f,0.f,0.f,0.f,0.f,0.f,0.f,0.f};
    v8f acc01 = acc00, acc10 = acc00, acc11 = acc00;

    for (int k0 = 0; k0 < IN_DIM; k0 += 64) {
        if (k0 + 64 < IN_DIM) {
            __builtin_prefetch(aRow0 + k0 + 64, 0, 3);   // global_prefetch_b8 (WGP)
            __builtin_prefetch(aRow1 + k0 + 64, 0, 3);
        }
#pragma unroll
        for (int kk = 0; kk < 64; kk += 4) {
            const int k = k0 + kk;
            v2f a0 = *(const v2f*)(aRow0 + k);           // {A[M][k+2h], A[M][k+2h+1]}
            v2f a1 = *(const v2f*)(aRow1 + k);
            v2f b0, b1;
            b0.x = bCol0[(size_t)k * NCOL];              // B[k+2h  ][N]
            b0.y = bCol0[(size_t)(k + 1) * NCOL];        // B[k+2h+1][N]
            b1.x = bCol1[(size_t)k * NCOL];
            b1.y = bCol1[(size_t)(k + 1) * NCOL];
            acc00 = __builtin_amdgcn_wmma_f32_16x16x4_f32(false, a0, false, b0,
                                                          (short)0, acc00, false, false);
            acc01 = __builtin_amdgcn_wmma_f32_16x16x4_f32(false, a0, false, b1,
                                                          (short)0, acc01, false, false);
            acc10 = __builtin_amdgcn_wmma_f32_16x16x4_f32(false, a1, false, b0,
                                                          (short)0, acc10, false, false);
            acc11 = __builtin_amdgcn_wmma_f32_16x16x4_f32(false, a1, false, b1,
                                                          (short)0, acc11, false, false);
        }
    }

    float* out00 = u + (size_t)row0 * NCOL + col0 + l16;
    float* out01 = out00 + 16;
    float* out10 = out00 + (size_t)16 * NCOL;
    float* out11 = out10 + 16;
#pragma unroll
    for (int r = 0; r < 8; ++r) {
        const size_t roff = (size_t)(r + half * 8) * NCOL;
        out00[roff] = acc00[r];
        out01[roff] = acc01[r];
        out10[roff] = acc10[r];
        out11[roff] = acc11[r];
    }
}

// ---------------------------------------------------------------------------
// Kernel 2: the diagonal-recurrence LSTM scan. One thread per state channel,
// 16 single-wave blocks spread across WGPs (512 independent chains).
// ---------------------------------------------------------------------------
__device__ __forceinline__ float sigmoid_f(float v) {
    return 1.f / (1.f + __expf(-v));
}
__device__ __forceinline__ float tanh_f(float v) {
    return 1.f - 2.f / (__expf(2.f * v) + 1.f);
}

__global__ void scan_kernel(const float* __restrict__ u,
                            const float* __restrict__ w_rec,
                            const float* __restrict__ bias,
                            float* __restrict__ h_n, float* __restrict__ c_n) {
    const int s = blockIdx.x * blockDim.x + threadIdx.x;
    const float wf = w_rec[s],             wi = w_rec[S_DIM + s];
    const float wo = w_rec[2 * S_DIM + s], wg = w_rec[3 * S_DIM + s];
    const float bf = bias[s],              bi = bias[S_DIM + s];
    const float bo = bias[2 * S_DIM + s],  bg = bias[3 * S_DIM + s];

    float h = 0.f, c = 0.f;
    h_n[s] = 0.f;                // row 0 of h_n / c_n
    c_n[s] = 0.f;

    const float* up = u + s;
    for (int t = 0; t < T_DIM; ++t) {
        const float* ut = up + (size_t)t * NCOL;
        float f = sigmoid_f(bf + wf * h + ut[0]);
        float i = sigmoid_f(bi + wi * h + ut[S_DIM]);
        float o = sigmoid_f(bo + wo * h + ut[2 * S_DIM]);
        float g = tanh_f   (bg + wg * h + ut[3 * S_DIM]);
        c = f * c + i * g;
        h = o * tanh_f(c);
        h_n[(size_t)(t + 1) * S_DIM + s] = h;
        c_n[(size_t)(t + 1) * S_DIM + s] = c;
    }
}

// ---------------------------------------------------------------------------
// Kernel 3: q[t] = dot(h_n[t+1], W_reg) + b_reg. One wave per output row,
// wave32 shuffle-xor reduction.
// ---------------------------------------------------------------------------
__global__ void q_kernel(const float* __restrict__ h_n,
                         const float* __restrict__ w_reg,
                         const float* __restrict__ b_reg,
                         float* __restrict__ q) {
    const int lane = threadIdx.x & 31;
    const int t = (blockIdx.x * blockDim.x + threadIdx.x) >> 5;
    const float* hr = h_n + (size_t)(t + 1) * S_DIM;
    float acc = 0.f;
#pragma unroll
    for (int j = 0; j < S_DIM; j += 32)
        acc += hr[j + lane] * w_reg[j + lane];
#pragma unroll
    for (int off = 16; off; off >>= 1)
        acc += __shfl_xor(acc, off, 32);
    if (lane == 0) q[t] = acc + b_reg[0];
}

// ---------------------------------------------------------------------------
extern "C" void kernel_launch(void* const* d_in, const int* in_sizes, int n_in,
                              void* d_out, int out_size, void* d_ws, size_t ws_size,
                              hipStream_t stream) {
    (void)in_sizes; (void)n_in; (void)out_size; (void)ws_size;
    const float* x     = (const float*)d_in[0];
    const float* w_in  = (const float*)d_in[1];
    const float* w_rec = (const float*)d_in[2];
    const float* bias  = (const float*)d_in[3];
    const float* w_reg = (const float*)d_in[4];
    const float* b_reg = (const float*)d_in[5];

    float* out = (float*)d_out;
    float* q   = out;                                       // (T)
    float* h_n = out + T_DIM;                               // (T+1, S)
    float* c_n = h_n + (size_t)(T_DIM + 1) * S_DIM;         // (T+1, S)

    float* u  = (float*)d_ws;                               // (T, NCOL)  134 MB
    float* bt = u + (size_t)T_DIM * NCOL;                   // (IN, NCOL) 8.4 MB

    wtrans_kernel<<<(IN_DIM * NCOL) / 256, 256, 0, stream>>>(w_in, bt);

    dim3 ggrid(NCOL / 256, T_DIM / 32);   // 8 waves/block, 32x32 tile per wave
    gemm_u_kernel<<<ggrid, 256, 0, stream>>>(x, bt, u);

    scan_kernel<<<S_DIM / 32, 32, 0, stream>>>(u, w_rec, bias, h_n, c_n);

    q_kernel<<<(T_DIM * 32) / 256, 256, 0, stream>>>(h_n, w_reg, b_reg, q);
}